// CQAttention_3968549781690
// MI455X (gfx1250) — compile-verified
//
#include <hip/hip_runtime.h>
#include <hip/hip_bf16.h>

// CQAttention for MI455X (gfx1250, wave32).
// GEMM stages: V_WMMA_F32_16X16X4_F32 (fp32 reference => fp32 matrix path).
// LDS staging: GLOBAL_LOAD_ASYNC_TO_LDS_B32/B128 (ASYNCcnt) via inline asm.

typedef __attribute__((ext_vector_type(2))) float v2f;
typedef __attribute__((ext_vector_type(8))) float v8f;

#define NEG_INF_F (-1e30f)
constexpr int Bn = 64;    // batch
constexpr int Dd = 256;   // feature dim
constexpr int LC = 1024;  // context length
constexpr int LQ = 256;   // query length

__device__ __forceinline__ v8f wmma4(v2f a, v2f b, v8f c) {
  // D = A(16x4) * B(4x16) + C(16x16), fp32
  return __builtin_amdgcn_wmma_f32_16x16x4_f32(
      /*neg_a=*/false, a, /*neg_b=*/false, b,
      /*c_mod=*/(short)0, c, /*reuse_a=*/false, /*reuse_b=*/false);
}

// LDS byte offset = low 32 bits of the flat shared-aperture address
__device__ __forceinline__ uint32_t lds_off32(const void* p) {
  return (uint32_t)(uintptr_t)p;
}
// async copy global -> LDS, 16B per lane (ASYNCcnt)
__device__ __forceinline__ void async_ld_b128(uint32_t lds, const void* g) {
  asm volatile("global_load_async_to_lds_b128 %0, %1, off"
               :: "v"(lds), "v"(g) : "memory");
}
// async copy global -> LDS, 4B per lane (ASYNCcnt)
__device__ __forceinline__ void async_ld_b32(uint32_t lds, const void* g) {
  asm volatile("global_load_async_to_lds_b32 %0, %1, off"
               :: "v"(lds), "v"(g) : "memory");
}
__device__ __forceinline__ void wait_async0() {
  asm volatile("s_wait_asynccnt 0x0" ::: "memory");
}

// ---------------------------------------------------------------------------
// K1: row bias a[b,c] = Ct[b,c,:]·w1 ; col bias bb[b,q] = Qt[b,q,:]·w2
// ---------------------------------------------------------------------------
__global__ __launch_bounds__(256) void k_bias(const float* __restrict__ C,
                                              const float* __restrict__ Q,
                                              const float* __restrict__ w,
                                              float* __restrict__ a_row,
                                              float* __restrict__ b_col) {
  int blk = blockIdx.x;
  int b = blk / 5, part = blk % 5;
  int tid = threadIdx.x;
  if (part < 4) {
    int c = part * 256 + tid;
    const float* Cb = C + (size_t)b * Dd * LC + c;
    float acc = 0.f;
#pragma unroll 4
    for (int d = 0; d < Dd; ++d) acc += Cb[(size_t)d * LC] * w[d];
    a_row[b * LC + c] = acc;
  } else {
    int q = tid;
    const float* Qb = Q + (size_t)b * Dd * LQ + q;
    float acc = 0.f;
#pragma unroll 4
    for (int d = 0; d < Dd; ++d) acc += Qb[(size_t)d * LQ] * w[Dd + d];
    b_col[b * LQ + q] = acc;
  }
}

// ---------------------------------------------------------------------------
// K2: M[b,c,q] = sum_d C[b,d,c]*w3[d]*Q[b,d,q]   (WMMA, K-loop over D)
// workgroup = (b, 64-row c slab). Each wave: 16 c-rows x 128 q columns held
// as 8 accumulators; the A-fragment is loaded once per K-step and shared by
// all 8 WMMAs (2 A + 16 B loads per 8 WMMAs).
// ---------------------------------------------------------------------------
__global__ __launch_bounds__(256) void k_M(const float* __restrict__ C,
                                           const float* __restrict__ Q,
                                           const float* __restrict__ w,
                                           float* __restrict__ M) {
  __shared__ float w3s[Dd];
  int b = blockIdx.x >> 4;
  int cbase = (blockIdx.x & 15) * 64;
  int tid = threadIdx.x;
  w3s[tid] = w[2 * Dd + tid];
  __syncthreads();

  int wave = tid >> 5, lane = tid & 31;
  const float* Cb = C + (size_t)b * Dd * LC;
  const float* Qb = Q + (size_t)b * Dd * LQ;
  float* Mb = M + (size_t)b * LC * LQ;

  int cw = cbase + (wave & 3) * 16;     // this wave's 16 c-rows
  int row = cw + (lane & 15);           // A-frag: lane -> matrix row (c)
  int khi = (lane >> 4) * 2;            // lanes 16-31 hold K=2,3
  int qbase = (wave >> 2) * 128;        // this wave's 128 q columns
  int qn = qbase + (lane & 15);         // B-frag: lane -> column (q)

  v8f acc[8] = {};
  for (int d0 = 0; d0 < Dd; d0 += 4) {
    int ka = d0 + khi;
    v2f af;
    af.x = Cb[(size_t)ka * LC + row] * w3s[ka];
    af.y = Cb[(size_t)(ka + 1) * LC + row] * w3s[ka + 1];
    const float* q0p = Qb + (size_t)ka * LQ + qn;
    const float* q1p = Qb + (size_t)(ka + 1) * LQ + qn;
#pragma unroll
    for (int t = 0; t < 8; ++t) {
      v2f bf;
      bf.x = q0p[t * 16];
      bf.y = q1p[t * 16];
      acc[t] = wmma4(af, bf, acc[t]);
    }
  }
  // D layout: v[i]: lanes0-15 -> (c=cw+i, q), lanes16-31 -> (c=cw+8+i, q)
  int cst = cw + (lane >> 4) * 8;
#pragma unroll
  for (int t = 0; t < 8; ++t) {
    int qc = qbase + t * 16 + (lane & 15);
#pragma unroll
    for (int i = 0; i < 8; ++i) Mb[(size_t)(cst + i) * LQ + qc] = acc[t][i];
  }
}

// ---------------------------------------------------------------------------
// K3: per-(b,c) softmax stats over q (for S1). wave per row, shuffle reduce.
// ---------------------------------------------------------------------------
__global__ __launch_bounds__(256) void k_rowstats(const float* __restrict__ M,
                                                  const float* __restrict__ a_row,
                                                  const float* __restrict__ b_col,
                                                  const float* __restrict__ q_mask,
                                                  float* __restrict__ rmax,
                                                  float* __restrict__ rsum) {
  int tid = threadIdx.x;
  int wave = tid >> 5, lane = tid & 31;
  int rowid = blockIdx.x * 8 + wave;   // rowid = b*LC + c
  int b = rowid / LC;
  const float* Mr = M + (size_t)rowid * LQ;
  float ac = a_row[rowid];
  float vals[8];
  float m = NEG_INF_F;
#pragma unroll
  for (int i = 0; i < 8; ++i) {
    int q = i * 32 + lane;
    float qm = q_mask[b * LQ + q];
    float s = ac + b_col[b * LQ + q] + Mr[q];
    float v = s * qm + (1.f - qm) * NEG_INF_F;
    vals[i] = v;
    m = fmaxf(m, v);
  }
#pragma unroll
  for (int off = 16; off >= 1; off >>= 1) m = fmaxf(m, __shfl_xor(m, off));
  float sum = 0.f;
#pragma unroll
  for (int i = 0; i < 8; ++i) sum += __expf(vals[i] - m);
#pragma unroll
  for (int off = 16; off >= 1; off >>= 1) sum += __shfl_xor(sum, off);
  if (lane == 0) { rmax[rowid] = m; rsum[rowid] = sum; }
}

// ---------------------------------------------------------------------------
// K4: per-(b,q) softmax stats over c (for S2). Online softmax, coalesced scan.
// ---------------------------------------------------------------------------
__global__ __launch_bounds__(256) void k_colstats(const float* __restrict__ M,
                                                  const float* __restrict__ a_row,
                                                  const float* __restrict__ b_col,
                                                  const float* __restrict__ c_mask,
                                                  float* __restrict__ cmax,
                                                  float* __restrict__ csum) {
  int b = blockIdx.x;
  int q = threadIdx.x;
  float bq = b_col[b * LQ + q];
  const float* Mb = M + (size_t)b * LC * LQ + q;
  float m = NEG_INF_F, s = 0.f;
  for (int c = 0; c < LC; ++c) {
    float cmv = c_mask[b * LC + c];
    float v = (a_row[b * LC + c] + bq + Mb[(size_t)c * LQ]) * cmv +
              (1.f - cmv) * NEG_INF_F;
    float mn = fmaxf(m, v);
    s = s * __expf(m - mn) + __expf(v - mn);
    m = mn;
  }
  cmax[b * LQ + q] = m;
  csum[b * LQ + q] = s;
}

// ---------------------------------------------------------------------------
// K5: materialize S1 (in place over M) and S2 (separate buffer)
// ---------------------------------------------------------------------------
__global__ __launch_bounds__(256) void k_s1s2(float* __restrict__ M,
                                              float* __restrict__ S2,
                                              const float* __restrict__ a_row,
                                              const float* __restrict__ b_col,
                                              const float* __restrict__ c_mask,
                                              const float* __restrict__ q_mask,
                                              const float* __restrict__ rmax,
                                              const float* __restrict__ rsum,
                                              const float* __restrict__ cmax,
                                              const float* __restrict__ csum) {
  int rowid = blockIdx.x;  // b*LC + c
  int b = rowid / LC;
  int q = threadIdx.x;
  size_t idx = (size_t)rowid * LQ + q;
  float s = a_row[rowid] + b_col[b * LQ + q] + M[idx];
  float qm = q_mask[b * LQ + q];
  float cmv = c_mask[rowid];
  float v1 = s * qm + (1.f - qm) * NEG_INF_F;
  float v2 = s * cmv + (1.f - cmv) * NEG_INF_F;
  M[idx] = __expf(v1 - rmax[rowid]) / rsum[rowid];
  S2[idx] = __expf(v2 - cmax[b * LQ + q]) / csum[b * LQ + q];
}

// ---------------------------------------------------------------------------
// K6: T[b,q,d] = sum_c S2[b,c,q] * C[b,d,c]   (WMMA, K-loop over LC)
// workgroup = (b, 16-wide d slab); C slab staged in LDS via async B128
// (pitch 524: 16B-aligned rows AND conflict-free 16-lane fragment reads).
// B-fragment is hoisted and shared by the wave's two q-tile accumulators.
// ---------------------------------------------------------------------------
__global__ __launch_bounds__(256) void k_T(const float* __restrict__ C,
                                           const float* __restrict__ S2,
                                           float* __restrict__ T) {
  __shared__ __align__(16) float ldsC[16 * 524];
  int b = blockIdx.x >> 4;
  int d0 = (blockIdx.x & 15) * 16;
  int tid = threadIdx.x;
  int wave = tid >> 5, lane = tid & 31;
  const float* Cb = C + (size_t)b * Dd * LC;
  const float* S2b = S2 + (size_t)b * LC * LQ;
  int khi = (lane >> 4) * 2;
  int nl = lane & 15;

  v8f acc[2] = {};
  for (int chunk = 0; chunk < 2; ++chunk) {
    __syncthreads();
    // async stage: 16 rows x 512 floats = 2048 B128 transfers (8/thread)
#pragma unroll
    for (int j = 0; j < 8; ++j) {
      int fid = (tid + j * 256) * 4;       // dword index within slab
      int d = fid >> 9, cc = fid & 511;
      async_ld_b128(lds_off32(&ldsC[d * 524 + cc]),
                    Cb + (size_t)(d0 + d) * LC + chunk * 512 + cc);
    }
    wait_async0();
    __syncthreads();
    for (int c0 = 0; c0 < 512; c0 += 4) {
      v2f bf;                                  // B[c, d] = Ct[c, d0+d]
      bf.x = ldsC[nl * 524 + c0 + khi];
      bf.y = ldsC[nl * 524 + c0 + khi + 1];
      int ca = chunk * 512 + c0 + khi;
#pragma unroll
      for (int t = 0; t < 2; ++t) {
        int qr = (wave * 2 + t) * 16 + nl;     // A-frag row = q
        v2f af;
        af.x = S2b[(size_t)ca * LQ + qr];      // A[q, c] = S2[c, q]
        af.y = S2b[(size_t)(ca + 1) * LQ + qr];
        acc[t] = wmma4(af, bf, acc[t]);
      }
    }
  }
  int ds = d0 + nl;
#pragma unroll
  for (int t = 0; t < 2; ++t) {
    int qs = (wave * 2 + t) * 16 + (lane >> 4) * 8;
#pragma unroll
    for (int i = 0; i < 8; ++i)
      T[((size_t)b * LQ + qs + i) * Dd + ds] = acc[t][i];
  }
}

// ---------------------------------------------------------------------------
// K7: A = S1@Qt, Bm = S1@T, out = concat(Ct, A, Ct*A, Ct*Bm) transposed.
// workgroup = (b, 16-wide d slab). Qt staged via async B32 (transposing),
// Tt via async B128 (contiguous along d); pitch 20 keeps B128 16B-aligned,
// fragment reads are lane-consecutive => conflict-free. Each S1 A-fragment
// feeds two WMMAs. Results transposed via per-wave LDS so (4D, LC) stores
// are contiguous along c.
// ---------------------------------------------------------------------------
__global__ __launch_bounds__(256) void k_out(const float* __restrict__ C,
                                             const float* __restrict__ Q,
                                             const float* __restrict__ S1,
                                             const float* __restrict__ T,
                                             float* __restrict__ out) {
  __shared__ __align__(16) float ldsQt[LQ * 20];  // [q][d_local], pitch 20
  __shared__ __align__(16) float ldsTt[LQ * 20];
  __shared__ float ldsO[8 * 2 * 16 * 17];         // per-wave A/B staging [c][d]
  int b = blockIdx.x >> 4;
  int d0 = (blockIdx.x & 15) * 16;
  int tid = threadIdx.x;
  int wave = tid >> 5, lane = tid & 31;

  // stage Qt tile: ldsQt[q][dl] = Q[b, d0+dl, q]  (transpose -> B32 async)
#pragma unroll 4
  for (int i = 0; i < 16; ++i)
    async_ld_b32(lds_off32(&ldsQt[tid * 20 + i]),
                 Q + ((size_t)b * Dd + d0 + i) * LQ + tid);
  // stage Tt tile: ldsTt[q][dl] = T[b, q, d0+dl]  (contiguous -> B128 async)
#pragma unroll
  for (int j = 0; j < 4; ++j) {
    int fid = tid + j * 256;                 // 0..1023
    int q = fid >> 2, dl0 = (fid & 3) * 4;
    async_ld_b128(lds_off32(&ldsTt[q * 20 + dl0]),
                  T + ((size_t)b * LQ + q) * Dd + d0 + dl0);
  }
  wait_async0();
  __syncthreads();

  int khi = (lane >> 4) * 2;
  int nl = lane & 15;
  float* oA = &ldsO[wave * 2 * 16 * 17];
  float* oB = oA + 16 * 17;
  size_t obase = (size_t)b * 4 * Dd * LC;

  for (int t = 0; t < 8; ++t) {            // uniform across all 8 waves
    int c0 = (wave * 8 + t) * 16;
    const float* S1r = S1 + ((size_t)b * LC + c0 + nl) * LQ;
    v8f accA = {}, accB = {};
#pragma unroll 4
    for (int q0 = 0; q0 < LQ; q0 += 4) {
      int qa = q0 + khi;
      v2f af = *(const v2f*)(S1r + qa);     // S1[c, qa..qa+1], contiguous
      v2f bq, bt;
      bq.x = ldsQt[qa * 20 + nl];
      bq.y = ldsQt[(qa + 1) * 20 + nl];
      bt.x = ldsTt[qa * 20 + nl];
      bt.y = ldsTt[(qa + 1) * 20 + nl];
      accA = wmma4(af, bq, accA);           // A   accumulation
      accB = wmma4(af, bt, accB);           // Bm  accumulation
    }
    // stage D-fragments [c][d] for transpose
    int cst = (lane >> 4) * 8;
#pragma unroll
    for (int i = 0; i < 8; ++i) {
      oA[(cst + i) * 17 + nl] = accA[i];
      oB[(cst + i) * 17 + nl] = accB[i];
    }
    __syncthreads();
    // stores contiguous along c (lane -> c); two 64B segments per instr
#pragma unroll
    for (int r = 0; r < 8; ++r) {
      int dloc = (lane >> 4) * 8 + r;
      float aV = oA[nl * 17 + dloc];
      float bV = oB[nl * 17 + dloc];
      float ct = C[((size_t)b * Dd + d0 + dloc) * LC + c0 + nl];
      int dd = d0 + dloc;
      size_t col = (size_t)c0 + nl;
      out[obase + (size_t)dd * LC + col] = ct;                 // Ct
      out[obase + (size_t)(Dd + dd) * LC + col] = aV;          // A
      out[obase + (size_t)(2 * Dd + dd) * LC + col] = ct * aV; // Ct*A
      out[obase + (size_t)(3 * Dd + dd) * LC + col] = ct * bV; // Ct*Bm
    }
    __syncthreads();
  }
}

// ---------------------------------------------------------------------------
extern "C" void kernel_launch(void* const* d_in, const int* in_sizes, int n_in,
                              void* d_out, int out_size, void* d_ws, size_t ws_size,
                              hipStream_t stream) {
  (void)in_sizes; (void)n_in; (void)out_size; (void)ws_size;
  const float* C = (const float*)d_in[0];       // (B, D, LC)
  const float* Q = (const float*)d_in[1];       // (B, D, LQ)
  const float* c_mask = (const float*)d_in[2];  // (B, LC)
  const float* q_mask = (const float*)d_in[3];  // (B, LQ)
  const float* w = (const float*)d_in[4];       // (3D,)
  float* out = (float*)d_out;                   // (B, 4D, LC)

  float* wsf = (float*)d_ws;
  float* a_row = wsf;                           // B*LC
  float* b_col = a_row + (size_t)Bn * LC;       // B*LQ
  float* rmax = b_col + (size_t)Bn * LQ;        // B*LC
  float* rsum = rmax + (size_t)Bn * LC;         // B*LC
  float* cmax = rsum + (size_t)Bn * LC;         // B*LQ
  float* csum = cmax + (size_t)Bn * LQ;         // B*LQ
  float* Mbuf = csum + (size_t)Bn * LQ;         // B*LC*LQ (becomes S1)
  float* S2buf = Mbuf + (size_t)Bn * LC * LQ;   // B*LC*LQ
  float* Tbuf = S2buf + (size_t)Bn * LC * LQ;   // B*LQ*D

  k_bias<<<Bn * 5, 256, 0, stream>>>(C, Q, w, a_row, b_col);
  k_M<<<Bn * (LC / 64), 256, 0, stream>>>(C, Q, w, Mbuf);
  k_rowstats<<<Bn * LC / 8, 256, 0, stream>>>(Mbuf, a_row, b_col, q_mask, rmax, rsum);
  k_colstats<<<Bn, 256, 0, stream>>>(Mbuf, a_row, b_col, c_mask, cmax, csum);
  k_s1s2<<<Bn * LC, 256, 0, stream>>>(Mbuf, S2buf, a_row, b_col, c_mask, q_mask,
                                      rmax, rsum, cmax, csum);
  k_T<<<Bn * (Dd / 16), 256, 0, stream>>>(C, S2buf, Tbuf);
  k_out<<<Bn * (Dd / 16), 256, 0, stream>>>(C, Q, Mbuf, Tbuf, out);
}